// GCNClassifier_163208757523
// MI455X (gfx1250) — compile-verified
//
#include <hip/hip_runtime.h>

typedef __attribute__((ext_vector_type(2))) float v2f;
typedef __attribute__((ext_vector_type(8))) float v8f;

#define BATCH    32
#define SEQ      100
#define NTOK     (BATCH * SEQ)   // 3200
#define EMB_D    300
#define POS_D    30
#define NER_D    30
#define IN_D     360             // 300+30+30
#define MEM_D    200
#define NPAD     208             // 13 * 16
#define DEPREL_V 40

// ---------------------------------------------------------------------------
// Kernel 1: embedding gather + concat -> embs [NTOK, IN_D]
// ---------------------------------------------------------------------------
__global__ void k_gather(const int* __restrict__ words, const int* __restrict__ pos,
                         const int* __restrict__ ner, const float* __restrict__ emb,
                         const float* __restrict__ pos_emb, const float* __restrict__ ner_emb,
                         float* __restrict__ embs) {
  int n = blockIdx.x;          // token index
  int t = threadIdx.x;         // 0..383
  if (t >= IN_D) return;
  float v;
  if (t < EMB_D)               v = emb[words[n] * EMB_D + t];
  else if (t < EMB_D + POS_D)  v = pos_emb[pos[n] * POS_D + (t - EMB_D)];
  else                         v = ner_emb[ner[n] * NER_D + (t - EMB_D - POS_D)];
  embs[n * IN_D + t] = v;
}

// ---------------------------------------------------------------------------
// Kernel 2: denom[b,l] = 1 + #nonzero in adj row; mask = (rowsum+colsum == 0)
// ---------------------------------------------------------------------------
__global__ void k_denom_mask(const int* __restrict__ adj, float* __restrict__ denom,
                             float* __restrict__ mask_out) {
  int i = blockIdx.x * blockDim.x + threadIdx.x;   // b*SEQ + l
  if (i >= NTOK) return;
  int b = i / SEQ, l = i % SEQ;
  const int* base = adj + b * SEQ * SEQ;
  int rs = 0, cs = 0;
  for (int m = 0; m < SEQ; ++m) {
    rs += (base[l * SEQ + m] != 0);
    cs += (base[m * SEQ + l] != 0);
  }
  denom[i] = (float)rs + 1.0f;
  mask_out[i] = (rs + cs == 0) ? 1.0f : 0.0f;
}

// ---------------------------------------------------------------------------
// Kernel 3: pad weight matrix [N,K] -> [NPAD,K] (zero rows above N) + bias.
// ---------------------------------------------------------------------------
__global__ void k_pad(const float* __restrict__ W, const float* __restrict__ b,
                      float* __restrict__ Wpad, float* __restrict__ bpad,
                      int K, int N) {
  int i = blockIdx.x * blockDim.x + threadIdx.x;
  int total = NPAD * K;
  if (i < total) {
    int c = i / K, k = i % K;
    Wpad[i] = (c < N) ? W[c * K + k] : 0.0f;
  }
  if (i < NPAD) bpad[i] = (i < N) ? b[i] : 0.0f;
}

// ---------------------------------------------------------------------------
// Kernel 4: diagonal deprel aggregation.
// Ax[b,l,d] = sum_m deprel[adj[b,l,m]][d] * x[b,m,d]
// One workgroup per batch b. Stage x[b] (80KB) + deprel table (32KB) in LDS.
// ---------------------------------------------------------------------------
__global__ void k_aggregate(const int* __restrict__ adj, const float* __restrict__ deprel,
                            const float* __restrict__ x, float* __restrict__ Ax) {
  extern __shared__ float smem[];
  float* s_x   = smem;                       // SEQ * MEM_D
  float* s_dep = smem + SEQ * MEM_D;         // DEPREL_V * MEM_D
  int*   s_adj = (int*)(s_dep + DEPREL_V * MEM_D);   // SEQ
  int b = blockIdx.x;
  int t = threadIdx.x;                       // 0..255
  for (int i = t; i < SEQ * MEM_D; i += blockDim.x)      s_x[i]   = x[b * SEQ * MEM_D + i];
  for (int i = t; i < DEPREL_V * MEM_D; i += blockDim.x) s_dep[i] = deprel[i];
  __syncthreads();
  const int* arow = adj + b * SEQ * SEQ;
  for (int l = 0; l < SEQ; ++l) {
    if (t < SEQ) s_adj[t] = arow[l * SEQ + t];
    __syncthreads();
    if (t < MEM_D) {
      float acc = 0.0f;
      for (int m = 0; m < SEQ; ++m) {
        int idx = s_adj[m];   // deprel row 0 is all-zero (padding), no branch needed
        acc = fmaf(s_dep[idx * MEM_D + t], s_x[m * MEM_D + t], acc);
      }
      Ax[(b * SEQ + l) * MEM_D + t] = acc;
    }
    __syncthreads();
  }
}

// ---------------------------------------------------------------------------
// Kernel 5: WMMA GEMM  out = f( (A [+ A2]) @ Wpad^T + biasScale*biaspad )
//   RELUDIV: v = relu(v / denom[row]);  else identity.
// One wave per 16x16 output tile; K stepped by 4 via v_wmma_f32_16x16x4_f32.
// Wpad has NPAD (=208) rows -> B loads are unconditional; only the final
// store is guarded, keeping the K-loop branch-free (EXEC all-1s throughout).
// A layout (16x4 f32): lanes 0-15 hold K={0,1}, lanes 16-31 hold K={2,3}.
// C/D (16x16 f32): VGPR i -> M=i (lanes 0-15) / M=i+8 (lanes 16-31), N=lane&15.
// ---------------------------------------------------------------------------
template <int K, bool HAS_A2, bool RELUDIV>
__global__ void k_gemm_wmma(const float* __restrict__ A, const float* __restrict__ A2,
                            const float* __restrict__ Wpad, const float* __restrict__ biaspad,
                            float biasScale, const float* __restrict__ denom,
                            float* __restrict__ out, int M, int N) {
  constexpr int TILES_N = NPAD / 16;         // 13
  int totalTiles = (M >> 4) * TILES_N;
  int tile = blockIdx.x * (blockDim.x >> 5) + (threadIdx.x >> 5);
  if (tile >= totalTiles) return;            // wave-uniform exit

  int tm = tile / TILES_N;
  int tn = tile % TILES_N;
  int lane = threadIdx.x & 31;
  int half = lane >> 4;                      // 0 or 1
  int lid  = lane & 15;
  int arow = tm * 16 + lid;                  // M divisible by 16 -> always valid
  int bcol = tn * 16 + lid;                  // < NPAD, always valid for Wpad

  const float* pA  = A + (size_t)arow * K + 2 * half;
  const float* pA2 = HAS_A2 ? (A2 + (size_t)arow * K + 2 * half) : nullptr;
  const float* pB  = Wpad + (size_t)bcol * K + 2 * half;

  v8f acc = {};
#pragma unroll 10
  for (int k = 0; k < K; k += 4) {
    v2f a = *(const v2f*)(pA + k);
    if (HAS_A2) a += *(const v2f*)(pA2 + k);
    v2f bv = *(const v2f*)(pB + k);
    acc = __builtin_amdgcn_wmma_f32_16x16x4_f32(
        /*neg_a=*/false, a, /*neg_b=*/false, bv,
        /*c_mod=*/(short)0, acc, /*reuse_a=*/false, /*reuse_b=*/false);
  }

  float bval = biaspad[bcol] * biasScale;
  bool storeok = (bcol < N);
#pragma unroll
  for (int i = 0; i < 8; ++i) {
    int m = tm * 16 + i + half * 8;
    float v = acc[i] + bval;
    if (RELUDIV) v = fmaxf(v / denom[m], 0.0f);
    if (storeok) out[(size_t)m * N + bcol] = v;
  }
}

// ---------------------------------------------------------------------------
extern "C" void kernel_launch(void* const* d_in, const int* in_sizes, int n_in,
                              void* d_out, int out_size, void* d_ws, size_t ws_size,
                              hipStream_t stream) {
  (void)in_sizes; (void)n_in; (void)out_size; (void)ws_size;
  const int*   adj     = (const int*)d_in[0];
  const int*   words   = (const int*)d_in[1];
  const int*   pos     = (const int*)d_in[2];
  const int*   ner     = (const int*)d_in[3];
  const float* emb     = (const float*)d_in[4];
  const float* pos_emb = (const float*)d_in[5];
  const float* ner_emb = (const float*)d_in[6];
  const float* deprel  = (const float*)d_in[7];
  const float* Wp      = (const float*)d_in[8];
  const float* bp      = (const float*)d_in[9];
  const float* W0      = (const float*)d_in[10];
  const float* b0      = (const float*)d_in[11];
  const float* W1      = (const float*)d_in[12];
  const float* b1      = (const float*)d_in[13];

  float* out  = (float*)d_out;                 // x [NTOK*MEM_D] then mask [NTOK]
  float* mask = out + (size_t)NTOK * MEM_D;

  float* ws     = (float*)d_ws;
  float* embs   = ws;                                   // NTOK*IN_D      = 1,152,000
  float* x0     = embs   + (size_t)NTOK * IN_D;         // NTOK*MEM_D     =   640,000
  float* x1     = x0     + (size_t)NTOK * MEM_D;
  float* Ax     = x1     + (size_t)NTOK * MEM_D;
  float* denom  = Ax     + (size_t)NTOK * MEM_D;        // NTOK
  float* Wp_pad = denom  + NTOK;                        // NPAD*IN_D
  float* W0_pad = Wp_pad + (size_t)NPAD * IN_D;         // NPAD*MEM_D
  float* W1_pad = W0_pad + (size_t)NPAD * MEM_D;
  float* bp_pad = W1_pad + (size_t)NPAD * MEM_D;        // NPAD each
  float* b0_pad = bp_pad + NPAD;
  float* b1_pad = b0_pad + NPAD;

  // 0. pad weights/biases to NPAD columns (rows of W) so GEMM loads are unguarded
  k_pad<<<(NPAD * IN_D  + 255) / 256, 256, 0, stream>>>(Wp, bp, Wp_pad, bp_pad, IN_D,  MEM_D);
  k_pad<<<(NPAD * MEM_D + 255) / 256, 256, 0, stream>>>(W0, b0, W0_pad, b0_pad, MEM_D, MEM_D);
  k_pad<<<(NPAD * MEM_D + 255) / 256, 256, 0, stream>>>(W1, b1, W1_pad, b1_pad, MEM_D, MEM_D);

  // 1. gather/concat embeddings
  k_gather<<<NTOK, 384, 0, stream>>>(words, pos, ner, emb, pos_emb, ner_emb, embs);
  // 2. denom + mask (constant across layers; mask written straight to output)
  k_denom_mask<<<(NTOK + 255) / 256, 256, 0, stream>>>(adj, denom, mask);
  // 3. preprocessor: x0 = embs @ Wp^T + bp   (tiles = 200*13 = 2600, 4 waves/block)
  k_gemm_wmma<IN_D, false, false><<<650, 128, 0, stream>>>(
      embs, nullptr, Wp_pad, bp_pad, 1.0f, nullptr, x0, NTOK, MEM_D);

  size_t lds = (size_t)(SEQ * MEM_D + DEPREL_V * MEM_D) * sizeof(float) + SEQ * sizeof(int);
  // layer 0: x1 = relu(((Ax + x0) @ W0^T + 2*b0) / denom)
  k_aggregate<<<BATCH, 256, lds, stream>>>(adj, deprel, x0, Ax);
  k_gemm_wmma<MEM_D, true, true><<<650, 128, 0, stream>>>(
      Ax, x0, W0_pad, b0_pad, 2.0f, denom, x1, NTOK, MEM_D);
  // layer 1 -> final output
  k_aggregate<<<BATCH, 256, lds, stream>>>(adj, deprel, x1, Ax);
  k_gemm_wmma<MEM_D, true, true><<<650, 128, 0, stream>>>(
      Ax, x1, W1_pad, b1_pad, 2.0f, denom, out, NTOK, MEM_D);
}